// ChromaticPositionalEncoding_8358006358358
// MI455X (gfx1250) — compile-verified
//
#include <hip/hip_runtime.h>
#include <cstdint>

// ChromaticPositionalEncoding, MI455X (gfx1250).
// Memory-bound elementwise add: 472 MB mandatory HBM traffic, ~20us floor at
// 23.3 TB/s. No matrix ops -> no WMMA; the CDNA5 feature path exercised here is
// the async copy engine (global_load_async_to_lds_b128 / ASYNCcnt /
// s_wait_asynccnt), global_prefetch_b8, and NT-hinted b128 streaming.

typedef __attribute__((ext_vector_type(4))) float v4f;

#define B_   128
#define H_   30
#define W_   30
#define D_   512
#define NCOL 10
#define HALF_F4 (D_ / 2 / 4)          // 64 v4f per half-feature
#define PIX_F4  (D_ / 4)              // 128 v4f per pixel
#define ROW_F4  (W_ * PIX_F4)         // 3840 v4f per (b,h) row
#define NTHREADS 256

// Async DMA: global -> LDS, 16B per active lane, tracked on ASYNCcnt.
// Syntax: global_load_async_to_lds_b128 vdst(LDS byte addr), vaddr(u32 offset), saddr(64-bit base)
__device__ __forceinline__ void async_copy_b128(uint32_t lds_addr,
                                                const void* gbase,
                                                uint32_t byte_off) {
    asm volatile("global_load_async_to_lds_b128 %0, %1, %2"
                 :: "v"(lds_addr), "v"(byte_off), "s"(gbase)
                 : "memory");
}

__device__ __forceinline__ void wait_async_all() {
    asm volatile("s_wait_asynccnt 0" ::: "memory");
}

__global__ __launch_bounds__(NTHREADS)
void ChromaticPositionalEncoding_8358006358358_kernel(
    const float* __restrict__ x,
    const int*   __restrict__ color_idx,
    const float* __restrict__ spatial_pe,
    const float* __restrict__ chromatic_pe,
    float*       __restrict__ out) {
    // LDS staging: 30 KB spatial slice + 10 KB chromatic table + indices (~41 KB of 320 KB WGP pool)
    __shared__ v4f s_spat[W_ * HALF_F4];     // spatial_pe[h, 0..29, :]
    __shared__ v4f s_chrom[NCOL * HALF_F4];  // full chromatic table
    __shared__ int s_idx[W_];

    const int tid = threadIdx.x;
    const int row = blockIdx.x;       // (b,h) flattened: row = b*H + h
    const int h   = row % H_;

    const v4f* __restrict__ xrow = reinterpret_cast<const v4f*>(x)  + (size_t)row * ROW_F4;
    v4f*       __restrict__ orow = reinterpret_cast<v4f*>(out)      + (size_t)row * ROW_F4;
    const void* srow  = (const void*)(spatial_pe + (size_t)h * (W_ * D_ / 2));
    const void* cbase = (const void*)chromatic_pe;
    const int* __restrict__ irow = color_idx + (size_t)row * W_;

    // Stage chromatic table: 640 v4f via async DMA (reused ~11520x overall).
    for (int i = tid; i < NCOL * HALF_F4; i += NTHREADS) {
        async_copy_b128((uint32_t)(uintptr_t)&s_chrom[i], cbase, (uint32_t)(i * 16));
    }
    // Stage this h's spatial slice: 1920 v4f via async DMA.
    for (int i = tid; i < W_ * HALF_F4; i += NTHREADS) {
        async_copy_b128((uint32_t)(uintptr_t)&s_spat[i], srow, (uint32_t)(i * 16));
    }
    if (tid < W_) s_idx[tid] = irow[tid];

    // Warm the cache hierarchy for this block's x row while the DMA drains
    // (one prefetch per 256B region; emits global_prefetch_b8, no counter).
    for (int i = tid * 16; i < ROW_F4; i += NTHREADS * 16) {
        __builtin_prefetch(xrow + i, 0, 1);
    }

    wait_async_all();      // s_wait_asynccnt 0: this wave's DMA done
    __syncthreads();       // all 8 waves' staging visible in LDS

    // Streaming add: 15 b128 load+store pairs per thread, fully coalesced.
    // Within a wave32, q spans 32 consecutive values -> the spatial/chromatic
    // branch and s_idx[w] are wave-uniform; LDS reads are conflict-free.
#pragma unroll
    for (int k = 0; k < ROW_F4 / NTHREADS; ++k) {
        const int i = k * NTHREADS + tid;
        const int w = i >> 7;          // pixel within row (128 v4f/pixel)
        const int q = i & 127;         // v4f within pixel

        v4f pe;
        if (q < HALF_F4) {
            pe = s_spat[(w << 6) + q];
        } else {
            pe = s_chrom[(s_idx[w] << 6) + (q - HALF_F4)];
        }

        // x/out are touched exactly once: non-temporal to keep L2 for PE reuse.
        v4f xv = __builtin_nontemporal_load(xrow + i);
        v4f r  = xv + pe;
        __builtin_nontemporal_store(r, orow + i);
    }
}

extern "C" void kernel_launch(void* const* d_in, const int* in_sizes, int n_in,
                              void* d_out, int out_size, void* d_ws, size_t ws_size,
                              hipStream_t stream) {
    const float* x          = (const float*)d_in[0];
    const int*   color_idx  = (const int*)  d_in[1];
    const float* spatial_pe = (const float*)d_in[2];
    const float* chromatic  = (const float*)d_in[3];
    float*       out        = (float*)d_out;

    dim3 grid(B_ * H_);        // 3840 blocks (one per (b,h) row) = 30720 waves
    dim3 block(NTHREADS);      // 8 wave32s
    ChromaticPositionalEncoding_8358006358358_kernel<<<grid, block, 0, stream>>>(
        x, color_idx, spatial_pe, chromatic, out);
}